// CUDATropicalLinear_23295902613821
// MI455X (gfx1250) — compile-verified
//
#include <hip/hip_runtime.h>
#include <stdint.h>

// Tropical (max-plus) linear: y[b,i] = max_j (W[i,j] + x[b,j]) + bias[i]
// B=512, IN=1024, OUT=1024, all fp32.

#define BATCH 512
#define INF_  1024
#define OUTF  1024

#define BM 64
#define BN 64
#define BK 64
#define PAD 4
#define LDK (BK + PAD)       // 68 floats per row: 272B, 16B aligned, 4-bank row skew
#define NTHREADS 256

typedef __attribute__((ext_vector_type(2))) float f32x2;
typedef __attribute__((ext_vector_type(4))) float f32x4;
typedef __attribute__((ext_vector_type(4))) int   i32x4;

#if defined(__gfx1250__) && __has_builtin(__builtin_amdgcn_global_load_async_to_lds_b128)
#define HAVE_ASYNC 1
#else
#define HAVE_ASYNC 0
#endif

#if HAVE_ASYNC
typedef __attribute__((address_space(1))) i32x4 gvec_t;
typedef __attribute__((address_space(3))) i32x4 lvec_t;

static __device__ __forceinline__ void async_copy_b128(const float* g, const float* l) {
    // Flat LDS addresses carry the LDS byte offset in their low 32 bits
    // (aperture base has zero low bits), so truncation == addrspacecast.
    __builtin_amdgcn_global_load_async_to_lds_b128(
        (gvec_t*)(uintptr_t)g,
        (lvec_t*)(uint32_t)(uintptr_t)l,
        0, 0);
}
static __device__ __forceinline__ void wait_async0() {
#if __has_builtin(__builtin_amdgcn_s_wait_asynccnt)
    __builtin_amdgcn_s_wait_asynccnt(0);
#else
    asm volatile("s_wait_asynccnt 0" ::: "memory");
#endif
}
#endif

__global__ __launch_bounds__(NTHREADS)
void tropical_linear_kernel(const float* __restrict__ x,
                            const float* __restrict__ w,
                            const float* __restrict__ bias,
                            float* __restrict__ out) {
    __shared__ __align__(16) float xs[2][BM * LDK];
    __shared__ __align__(16) float ws[2][BN * LDK];

    const int tid = threadIdx.x;
    const int tx  = tid & 15;    // n-group lane
    const int ty  = tid >> 4;    // m-group lane

    const int bn0 = (blockIdx.x & 15) * BN;   // 16 tiles along OUT
    const int bm0 = (blockIdx.x >> 4) * BM;   // 8 tiles along BATCH

    float acc[4][4];
#pragma unroll
    for (int i = 0; i < 4; ++i)
#pragma unroll
        for (int j = 0; j < 4; ++j) acc[i][j] = -__builtin_inff();

    // ---- tile loader: 64x64 floats per matrix = 1024 b128 quads, 4 per thread ----
    auto issue_loads = [&](int buf, int k0) {
#pragma unroll
        for (int i = 0; i < 4; ++i) {
            const int idx  = tid + i * NTHREADS;   // 0..1023
            const int row  = idx >> 4;             // 0..63
            const int col4 = (idx & 15) << 2;      // float column, multiple of 4
            const float* gx = x + (size_t)(bm0 + row) * INF_ + k0 + col4;
            const float* gw = w + (size_t)(bn0 + row) * INF_ + k0 + col4;
            float* lx = &xs[buf][row * LDK + col4];
            float* lw = &ws[buf][row * LDK + col4];
#if HAVE_ASYNC
            async_copy_b128(gx, lx);
            async_copy_b128(gw, lw);
#else
            *(f32x4*)lx = *(const f32x4*)gx;
            *(f32x4*)lw = *(const f32x4*)gw;
#endif
        }
    };

    // ---- compute one 64-deep k chunk from LDS ----
    auto compute = [&](int buf) {
        for (int kk = 0; kk < BK; kk += 4) {
            f32x4 xv[4], wv[4];
#pragma unroll
            for (int mi = 0; mi < 4; ++mi)
                xv[mi] = *(const f32x4*)&xs[buf][(ty + 16 * mi) * LDK + kk];
#pragma unroll
            for (int ni = 0; ni < 4; ++ni)
                wv[ni] = *(const f32x4*)&ws[buf][(tx + 16 * ni) * LDK + kk];
#pragma unroll
            for (int mi = 0; mi < 4; ++mi) {
#pragma unroll
                for (int ni = 0; ni < 4; ++ni) {
                    // 2x v_pk_add_f32 (4 adds) + 2x v_max3_num_f32 (4 max merges)
                    f32x4 t = xv[mi] + wv[ni];
                    float m = acc[mi][ni];
                    m = __builtin_fmaxf(__builtin_fmaxf(m, t.x), t.y);
                    m = __builtin_fmaxf(__builtin_fmaxf(m, t.z), t.w);
                    acc[mi][ni] = m;
                }
            }
        }
    };

    // ---- software pipeline: async load tile t+1 while computing tile t ----
    issue_loads(0, 0);
#if HAVE_ASYNC
    wait_async0();
#endif
    __syncthreads();

    int buf = 0;
    for (int k0 = BK; k0 < INF_; k0 += BK) {
        issue_loads(buf ^ 1, k0);
        compute(buf);
#if HAVE_ASYNC
        wait_async0();
#endif
        __syncthreads();
        buf ^= 1;
    }
    compute(buf);

    // ---- epilogue: add bias, write out (lanes cover consecutive n: coalesced) ----
#pragma unroll
    for (int mi = 0; mi < 4; ++mi) {
        const int m = bm0 + ty + 16 * mi;
#pragma unroll
        for (int ni = 0; ni < 4; ++ni) {
            const int n = bn0 + tx + 16 * ni;
            out[(size_t)m * OUTF + n] = acc[mi][ni] + bias[n];
        }
    }
}

extern "C" void kernel_launch(void* const* d_in, const int* in_sizes, int n_in,
                              void* d_out, int out_size, void* d_ws, size_t ws_size,
                              hipStream_t stream) {
    const float* x    = (const float*)d_in[0];
    const float* wgt  = (const float*)d_in[1];
    const float* bias = (const float*)d_in[2];
    float* out = (float*)d_out;

    // grid: (BATCH/BM) * (OUTF/BN) = 8 * 16 = 128 workgroups, 8 waves each
    dim3 grid((BATCH / BM) * (OUTF / BN));
    tropical_linear_kernel<<<grid, NTHREADS, 0, stream>>>(x, wgt, bias, out);
}